// LIF_9912784520006
// MI455X (gfx1250) — compile-verified
//
#include <hip/hip_runtime.h>
#include <cstdint>

// LIF scan: B=4096 rows, L=4096 sequential steps per row.
// Bandwidth floor: 128MB @ 23.3 TB/s ~ 5.7us; with 1 wave/SIMD the real
// limiter is the per-step dependent-VALU chain, shortened here to
// fma||cmp -> cndmask (2 levels) by deferring the reset into the next step.
// Data movement rides gfx1250 async global<->LDS DMA (ASYNCcnt), coalesced,
// double-buffered, with conflict-free padded LDS tiles.

#define LIF_B 4096
#define LIF_L 4096

constexpr int ROWS   = 32;            // rows per workgroup = one per wave32 lane
constexpr int TW     = 64;            // time-steps per tile
constexpr int STRIDE = TW + 4;        // padded LDS row stride (floats): 68 -> conflict-free b128
constexpr int NTILES = LIF_L / TW;    // 64
constexpr int AI     = (ROWS * TW) / (32 * 4);  // async b128 instructions per tile = 16

#if __has_builtin(__builtin_amdgcn_s_wait_asynccnt)
#define WAIT_ASYNC(n) do { __builtin_amdgcn_s_wait_asynccnt(n); \
                           asm volatile("" ::: "memory"); } while (0)
#else
#define WAIT_ASYNC(n) asm volatile("s_wait_asynccnt %0" :: "n"(n) : "memory")
#endif

#if __has_builtin(__builtin_amdgcn_s_wait_dscnt)
#define WAIT_DS(n) do { __builtin_amdgcn_s_wait_dscnt(n); \
                        asm volatile("" ::: "memory"); } while (0)
#else
#define WAIT_DS(n) asm volatile("s_wait_dscnt %0" :: "n"(n) : "memory")
#endif

__global__ __launch_bounds__(32)
void lif_scan_kernel(const float* __restrict__ I,
                     float* __restrict__ spikes,
                     int*   __restrict__ hard_lat,
                     float* __restrict__ soft_lat)
{
    __shared__ float lbuf[2][ROWS * STRIDE];   // input tiles  (double buffered)
    __shared__ float sbuf[2][ROWS * STRIDE];   // output tiles (double buffered)

    const int lane  = threadIdx.x;             // 0..31, owns one row
    const int row0  = blockIdx.x * ROWS;
    const int myrow = row0 + lane;

    // DMA lane mapping: instruction j moves rows 2j,2j+1; 16 lanes x 16B cover
    // one contiguous 256B row-segment -> fully coalesced.
    const int lrow = lane >> 4;                // row within pair
    const int lcol = (lane & 15) * 4;          // float offset within row

    const float* gIn  = I      + (size_t)row0 * LIF_L;
    float*       gOut = spikes + (size_t)row0 * LIF_L;

    // State: w = PRE-reset membrane potential of previous step; sPrev = spike
    // at previous step. Reset is folded into the next integration:
    //   w_t = sPrev ? x_t : fma(w_{t-1}, 0.95, x_t)
    float w = 0.0f;
    bool  sPrev = false;
    float cnt = 0.0f, tsum = 0.0f;
    int   first = LIF_L;

    auto issue_load = [&](int tile, int buf) {
        const unsigned lbase = (unsigned)(uintptr_t)&lbuf[buf][0]; // LDS addr = low 32 bits
#pragma unroll
        for (int j = 0; j < AI; ++j) {
            const int r = 2 * j + lrow;
            const uint64_t ga =
                (uint64_t)(uintptr_t)(gIn + (size_t)r * LIF_L + (size_t)tile * TW + lcol);
            const unsigned la = lbase + (unsigned)((r * STRIDE + lcol) * 4);
            asm volatile("global_load_async_to_lds_b128 %0, %1, off"
                         :: "v"(la), "v"(ga) : "memory");
        }
    };

    auto issue_store = [&](int tile, int buf) {
        const unsigned sbase = (unsigned)(uintptr_t)&sbuf[buf][0];
#pragma unroll
        for (int j = 0; j < AI; ++j) {
            const int r = 2 * j + lrow;
            const uint64_t ga =
                (uint64_t)(uintptr_t)(gOut + (size_t)r * LIF_L + (size_t)tile * TW + lcol);
            const unsigned la = sbase + (unsigned)((r * STRIDE + lcol) * 4);
            asm volatile("global_store_async_from_lds_b128 %0, %1, off"
                         :: "v"(ga), "v"(la) : "memory");
        }
    };

    issue_load(0, 0);   // prime the pipeline

    for (int k = 0; k < NTILES; ++k) {
        const int buf = k & 1;
        if (k + 1 < NTILES) {
            issue_load(k + 1, (k + 1) & 1);
            // <=AI outstanding proves (loads retire in order, stores in order
            // with stores): tile-k loads landed AND the store buffer being
            // reused next has fully drained.
            WAIT_ASYNC(AI);
        } else {
            WAIT_ASYNC(0);
        }

        const int t0 = k * TW;
#pragma unroll 4
        for (int tt = 0; tt < TW; tt += 4) {
            // padded stride 68 -> bank-conflict-free ds_load_b128 per half-wave
            const float4 in = *(const float4*)&lbuf[buf][lane * STRIDE + tt];
            const float x[4] = {in.x, in.y, in.z, in.w};
            float o[4];
#pragma unroll
            for (int u = 0; u < 4; ++u) {
                const int t = t0 + tt + u;
                // speculative integrate (no reset applied yet)
                const float p = fmaf(w, 0.95f, x[u]);
                // apply previous step's reset: chain is fma||cmp -> cndmask
                w = sPrev ? x[u] : p;
                const bool s = (w >= 1.0f);            // hard spike
                o[u]  = s ? 1.0f : 0.0f;
                cnt  += o[u];
                tsum  = fmaf(o[u], (float)t, tsum);
                first = (s && first == LIF_L) ? t : first;
                sPrev = s;
            }
            float4 sp; sp.x = o[0]; sp.y = o[1]; sp.z = o[2]; sp.w = o[3];
            *(float4*)&sbuf[buf][lane * STRIDE + tt] = sp;
        }

        WAIT_DS(0);              // ds_store results visible to the DMA engine
        issue_store(k, buf);     // drain spikes tile with coalesced async stores
    }
    // S_ENDPGM performs an implicit wait-idle, draining remaining async stores.

    hard_lat[myrow] = first;                    // int32 bits into output buffer
    soft_lat[myrow] = tsum / (cnt + 1e-6f);     // EPS matches reference
}

extern "C" void kernel_launch(void* const* d_in, const int* in_sizes, int n_in,
                              void* d_out, int out_size, void* d_ws, size_t ws_size,
                              hipStream_t stream) {
    (void)in_sizes; (void)n_in; (void)d_ws; (void)ws_size; (void)out_size;
    const float* I = (const float*)d_in[0];

    // d_out layout: spikes [B*L] f32, hard_latency [B] i32 (raw bits),
    // soft_latency [B] f32 — concatenated in reference return order.
    float* out    = (float*)d_out;
    float* spikes = out;
    int*   hard   = (int*)(out + (size_t)LIF_B * LIF_L);
    float* soft   = out + (size_t)LIF_B * LIF_L + LIF_B;

    dim3 grid(LIF_B / ROWS), block(32);
    lif_scan_kernel<<<grid, block, 0, stream>>>(I, spikes, hard, soft);
}